// ShellConv_49452253446434
// MI455X (gfx1250) — compile-verified
//
#include <hip/hip_runtime.h>
#include <math.h>

typedef __attribute__((ext_vector_type(2))) float v2f;
typedef __attribute__((ext_vector_type(8))) float v8f;

#define BB   8
#define NN   4096
#define MM   2048
#define KK   32
#define F1C  128
#define F2C  256
#define FMID 64
#define CIN  192
#define XC   768            // CIN * DIV
#define ROWS 2097152        // 4*B*M*K
#define BM   16384          // B*M
#define EPS  1e-5f

// stats layout (floats): [0..3] bn1 sum, [4..7] bn1 sumsq,
// [8..39] bn2 sum, [40..71] bn2 sumsq, [72..263] bn3 sum, [264..455] bn3 sumsq
#define ST_B1S 0
#define ST_B1Q 4
#define ST_B2S 8
#define ST_B2Q 40
#define ST_B3S 72
#define ST_B3Q 264

__device__ inline float wred(float v) {
    for (int o = 16; o > 0; o >>= 1) v += __shfl_down(v, o, 32);
    return v;
}

__device__ inline v8f wmma4(v2f a, v2f b, v8f c) {
    return __builtin_amdgcn_wmma_f32_16x16x4_f32(false, a, false, b, (short)0, c, false, false);
}

// ---------------- KNN + local coords + BN1 stats ----------------
__global__ __launch_bounds__(256) void k_knn(const float* __restrict__ pts,
                                             const float* __restrict__ qrs,
                                             int* __restrict__ idx,
                                             float* __restrict__ locl,
                                             float* __restrict__ stats) {
    __shared__ float4 sp[2048];
    int b = blockIdx.y;
    int m = blockIdx.x * 256 + threadIdx.x;
    const float* pb = pts + (size_t)b * NN * 3;
    const float* q  = qrs + ((size_t)b * MM + m) * 3;
    float qx = q[0], qy = q[1], qz = q[2];
    float qq = qx * qx + qy * qy + qz * qz;

    float bd[KK];
    int   bi[KK];
#pragma unroll
    for (int i = 0; i < KK; i++) { bd[i] = -3.4e38f; bi[i] = 0; }

    for (int t = 0; t < 2; t++) {
        for (int i = threadIdx.x; i < 2048; i += 256) {
            int n = t * 2048 + i;
            float x = pb[n * 3 + 0], y = pb[n * 3 + 1], z = pb[n * 3 + 2];
            sp[i] = make_float4(x, y, z, x * x + y * y + z * z);
        }
        __syncthreads();
        for (int i = 0; i < 2048; i++) {
            float4 p = sp[i];
            float d = 2.f * (qx * p.x + qy * p.y + qz * p.z) - qq - p.w;
            if (d > bd[KK - 1]) {
                int j = KK - 1;
                while (j > 0 && bd[j - 1] < d) { bd[j] = bd[j - 1]; bi[j] = bi[j - 1]; j--; }
                bd[j] = d; bi[j] = t * 2048 + i;
            }
        }
        __syncthreads();
    }

    float s0 = 0, s1 = 0, s2 = 0, q0 = 0, q1 = 0, q2 = 0;
    size_t base = ((size_t)b * MM + m) * KK;
#pragma unroll 4
    for (int k = 0; k < KK; k++) {
        int n = bi[k];
        idx[base + k] = n;
        float lx = qx - pb[n * 3 + 0];
        float ly = qy - pb[n * 3 + 1];
        float lz = qz - pb[n * 3 + 2];
        float* L = locl + (base + k) * 3;
        L[0] = lx; L[1] = ly; L[2] = lz;
        s0 += lx; s1 += ly; s2 += lz;
        q0 += lx * lx; q1 += ly * ly; q2 += lz * lz;
    }
    s0 = wred(s0); s1 = wred(s1); s2 = wred(s2);
    q0 = wred(q0); q1 = wred(q1); q2 = wred(q2);
    if ((threadIdx.x & 31) == 0) {
        atomicAdd(&stats[ST_B1S + 0], s0); atomicAdd(&stats[ST_B1S + 1], s1);
        atomicAdd(&stats[ST_B1S + 2], s2);
        atomicAdd(&stats[ST_B1Q + 0], q0); atomicAdd(&stats[ST_B1Q + 1], q1);
        atomicAdd(&stats[ST_B1Q + 2], q2);
    }
}

// ---------------- fold BN1 into dense1 ----------------
__global__ void k_fold1(const float* __restrict__ stats, const float* __restrict__ g1,
                        const float* __restrict__ b1, const float* __restrict__ W1,
                        const float* __restrict__ c1, float* __restrict__ W1p,
                        float* __restrict__ c1p) {
    const float inv = 1.f / (float)ROWS;
    int o = threadIdx.x;  // 32
    float acc = c1[o];
    for (int c = 0; c < 3; c++) {
        float mean = stats[ST_B1S + c] * inv;
        float var  = stats[ST_B1Q + c] * inv - mean * mean;
        float s = g1[c] * rsqrtf(var + EPS);
        float t = b1[c] - mean * s;
        float w = W1[c * 32 + o];
        W1p[c * 32 + o] = s * w;
        acc += t * w;
    }
    c1p[o] = acc;
}

// ---------------- BN2 stats (recompute h1 on the fly) ----------------
__global__ __launch_bounds__(256) void k_bn2stats(const float* __restrict__ locl,
                                                  const float* __restrict__ W1p,
                                                  const float* __restrict__ c1p,
                                                  float* __restrict__ stats) {
    __shared__ float w1s[96], c1s[32];
    if (threadIdx.x < 96) w1s[threadIdx.x] = W1p[threadIdx.x];
    if (threadIdx.x < 32) c1s[threadIdx.x] = c1p[threadIdx.x];
    __syncthreads();
    float s[32], qv[32];
#pragma unroll
    for (int o = 0; o < 32; o++) { s[o] = 0.f; qv[o] = 0.f; }
    for (size_t r = (size_t)blockIdx.x * 256 + threadIdx.x; r < (size_t)ROWS;
         r += (size_t)gridDim.x * 256) {
        float lx = locl[r * 3 + 0], ly = locl[r * 3 + 1], lz = locl[r * 3 + 2];
#pragma unroll
        for (int o = 0; o < 32; o++) {
            float v = c1s[o] + lx * w1s[o] + ly * w1s[32 + o] + lz * w1s[64 + o];
            v = fmaxf(v, 0.f);
            s[o] += v; qv[o] += v * v;
        }
    }
    int lane = threadIdx.x & 31;
#pragma unroll
    for (int o = 0; o < 32; o++) {
        float ss = wred(s[o]), qq = wred(qv[o]);
        if (lane == 0) {
            atomicAdd(&stats[ST_B2S + o], ss);
            atomicAdd(&stats[ST_B2Q + o], qq);
        }
    }
}

// ---------------- fold BN2 into dense2 ----------------
__global__ void k_fold2(const float* __restrict__ stats, const float* __restrict__ g2,
                        const float* __restrict__ b2, const float* __restrict__ W2,
                        const float* __restrict__ c2, float* __restrict__ W2p,
                        float* __restrict__ c2p) {
    const float inv = 1.f / (float)ROWS;
    int o = threadIdx.x;  // 64
    float acc = c2[o];
    for (int c = 0; c < 32; c++) {
        float mean = stats[ST_B2S + c] * inv;
        float var  = stats[ST_B2Q + c] * inv - mean * mean;
        float s = g2[c] * rsqrtf(var + EPS);
        float t = b2[c] - mean * s;
        float w = W2[c * 64 + o];
        W2p[c * 64 + o] = s * w;
        acc += t * w;
    }
    c2p[o] = acc;
}

// ---------------- dense2 (WMMA f32 16x16x4) + K-maxpool ----------------
// Block = 256 threads = 8 waves, covers 128 rows (= 4 queries * K=32).
__global__ __launch_bounds__(256) void k_dense2pool(const float* __restrict__ locl,
                                                    const float* __restrict__ W1p,
                                                    const float* __restrict__ c1p,
                                                    const float* __restrict__ W2p,
                                                    const float* __restrict__ c2p,
                                                    float* __restrict__ X) {
    __shared__ float w1s[96], c1s[32], c2s[64];
    __shared__ float w2s[2048];
    __shared__ float h1[128 * 32];
    int tid = threadIdx.x;
    if (tid < 96) w1s[tid] = W1p[tid];
    if (tid < 32) c1s[tid] = c1p[tid];
    if (tid < 64) c2s[tid] = c2p[tid];
    for (int i = tid; i < 2048; i += 256) w2s[i] = W2p[i];
    __syncthreads();

    {   // compute h1 tile: thread -> half a row
        int r = tid >> 1, half = (tid & 1) * 16;
        size_t gr = (size_t)blockIdx.x * 128 + r;
        float lx = locl[gr * 3 + 0], ly = locl[gr * 3 + 1], lz = locl[gr * 3 + 2];
#pragma unroll
        for (int o = 0; o < 16; o++) {
            int cc = half + o;
            h1[r * 32 + cc] =
                fmaxf(c1s[cc] + lx * w1s[cc] + ly * w1s[32 + cc] + lz * w1s[64 + cc], 0.f);
        }
    }
    __syncthreads();

    int w = tid >> 5, lane = tid & 31;
    int lm = lane & 15, lh = lane >> 4;
    int rb = w * 16;

    v2f a[8];
#pragma unroll
    for (int kb = 0; kb < 8; kb++) {
        const float* p = &h1[(rb + lm) * 32 + kb * 4 + lh * 2];
        v2f af; af[0] = p[0]; af[1] = p[1];
        a[kb] = af;
    }

    int bm = blockIdx.x * 4 + (w >> 1);
    int d  = (w & 1) * 2 + lh;   // pool-group index within query

#pragma unroll
    for (int nb = 0; nb < 4; nb++) {
        v8f acc = {0.f, 0.f, 0.f, 0.f, 0.f, 0.f, 0.f, 0.f};
#pragma unroll
        for (int kb = 0; kb < 8; kb++) {
            int kr = kb * 4 + lh * 2;
            v2f bf;
            bf[0] = w2s[kr * 64 + nb * 16 + lm];
            bf[1] = w2s[(kr + 1) * 64 + nb * 16 + lm];
            acc = wmma4(a[kb], bf, acc);
        }
        int col = nb * 16 + lm;
        float bias = c2s[col];
        // relu then pool-of-8: D-tile VGPR r holds rows r (lanes 0-15) / r+8 (lanes 16-31)
        float p = 0.f;
#pragma unroll
        for (int r = 0; r < 8; r++) p = fmaxf(p, acc[r] + bias);
        X[(size_t)bm * XC + col * 4 + d] = p;
    }
}

// ---------------- gather prev_features + K-maxpool ----------------
__global__ __launch_bounds__(128) void k_prevpool(const float* __restrict__ prev,
                                                  const int* __restrict__ idx,
                                                  float* __restrict__ X) {
    __shared__ int si[KK];
    int bm = blockIdx.x;
    int b = bm >> 11;  // M = 2048
    if (threadIdx.x < KK) si[threadIdx.x] = idx[(size_t)bm * KK + threadIdx.x];
    __syncthreads();
    int c = threadIdx.x;
    const float* pb = prev + (size_t)b * NN * F1C;
#pragma unroll
    for (int d = 0; d < 4; d++) {
        float mx = -3.4e38f;
#pragma unroll
        for (int s = 0; s < 8; s++) {
            int n = si[d * 8 + s];
            mx = fmaxf(mx, pb[(size_t)n * F1C + c]);
        }
        X[(size_t)bm * XC + (FMID + c) * 4 + d] = mx;
    }
}

// ---------------- BN3 stats over pooled features ----------------
__global__ __launch_bounds__(256) void k_bn3stats(const float* __restrict__ X,
                                                  float* __restrict__ stats) {
    float s[3] = {0.f, 0.f, 0.f}, q[3] = {0.f, 0.f, 0.f};
    for (int r = blockIdx.x; r < BM; r += gridDim.x) {
        const float* row = X + (size_t)r * XC;
#pragma unroll
        for (int i = 0; i < 3; i++) {
            float v = row[threadIdx.x + i * 256];
            s[i] += v; q[i] += v * v;
        }
    }
#pragma unroll
    for (int i = 0; i < 3; i++) {
        int j = threadIdx.x + i * 256;
        atomicAdd(&stats[ST_B3S + (j >> 2)], s[i]);
        atomicAdd(&stats[ST_B3Q + (j >> 2)], q[i]);
    }
}

// ---------------- fold BN3 into the conv weights ----------------
__global__ __launch_bounds__(256) void k_fold3(const float* __restrict__ stats,
                                               const float* __restrict__ g3,
                                               const float* __restrict__ b3,
                                               const float* __restrict__ Wc,
                                               const float* __restrict__ bc,
                                               float* __restrict__ Wcp,
                                               float* __restrict__ bcp) {
    __shared__ float red[256];
    const float inv = 1.f / (float)(BM * 4);
    int o = blockIdx.x;
    float acc = 0.f;
    for (int j = threadIdx.x; j < XC; j += 256) {
        int c = j >> 2;
        float mean = stats[ST_B3S + c] * inv;
        float var  = stats[ST_B3Q + c] * inv - mean * mean;
        float sc = g3[c] * rsqrtf(var + EPS);
        float tc = b3[c] - mean * sc;
        float wv = Wc[(size_t)o * XC + j];
        Wcp[(size_t)o * XC + j] = sc * wv;
        acc += tc * wv;
    }
    red[threadIdx.x] = acc; __syncthreads();
    for (int st = 128; st > 0; st >>= 1) {
        if (threadIdx.x < st) red[threadIdx.x] += red[threadIdx.x + st];
        __syncthreads();
    }
    if (threadIdx.x == 0) bcp[o] = bc[o] + red[0];
}

// ---------------- final GEMM (16384x768 @ 768x256) via WMMA + ReLU ----------------
__global__ __launch_bounds__(256) void k_gemm_out(const float* __restrict__ X,
                                                  const float* __restrict__ Wcp,
                                                  const float* __restrict__ bcp,
                                                  float* __restrict__ out) {
    __shared__ float xs[16 * XC];  // 48 KB
    int tid = threadIdx.x;
    size_t r0 = (size_t)blockIdx.x * 16;
    for (int i = tid; i < 16 * XC; i += 256) xs[i] = X[r0 * XC + i];
    __syncthreads();

    int w = tid >> 5, lane = tid & 31, lm = lane & 15, lh = lane >> 4;
    int n0 = w * 32 + lm, n1 = n0 + 16;
    const float* B0 = Wcp + (size_t)n0 * XC;
    const float* B1 = Wcp + (size_t)n1 * XC;

    v8f acc0 = {0.f, 0.f, 0.f, 0.f, 0.f, 0.f, 0.f, 0.f};
    v8f acc1 = {0.f, 0.f, 0.f, 0.f, 0.f, 0.f, 0.f, 0.f};
    for (int kb = 0; kb < XC / 4; kb++) {
        int ko = kb * 4 + lh * 2;
        v2f af; af[0] = xs[lm * XC + ko]; af[1] = xs[lm * XC + ko + 1];
        v2f b0; b0[0] = B0[ko]; b0[1] = B0[ko + 1];
        v2f b1; b1[0] = B1[ko]; b1[1] = B1[ko + 1];
        acc0 = wmma4(af, b0, acc0);
        acc1 = wmma4(af, b1, acc1);
    }
    float bias0 = bcp[n0], bias1 = bcp[n1];
#pragma unroll
    for (int r = 0; r < 8; r++) {
        size_t row = r0 + r + lh * 8;
        out[row * F2C + n0] = fmaxf(acc0[r] + bias0, 0.f);
        out[row * F2C + n1] = fmaxf(acc1[r] + bias1, 0.f);
    }
}

extern "C" void kernel_launch(void* const* d_in, const int* in_sizes, int n_in,
                              void* d_out, int out_size, void* d_ws, size_t ws_size,
                              hipStream_t stream) {
    const float* points  = (const float*)d_in[0];
    const float* queries = (const float*)d_in[1];
    const float* prev    = (const float*)d_in[2];
    const float* g1 = (const float*)d_in[3];
    const float* b1 = (const float*)d_in[4];
    const float* W1 = (const float*)d_in[5];
    const float* c1 = (const float*)d_in[6];
    const float* g2 = (const float*)d_in[7];
    const float* b2 = (const float*)d_in[8];
    const float* W2 = (const float*)d_in[9];
    const float* c2 = (const float*)d_in[10];
    const float* g3 = (const float*)d_in[11];
    const float* b3 = (const float*)d_in[12];
    const float* Wc = (const float*)d_in[13];
    const float* bc = (const float*)d_in[14];
    float* out = (float*)d_out;

    // workspace carve-up
    float* stats = (float*)d_ws;          // 512 floats
    float* W1p = stats + 512;             // 96
    float* c1p = W1p + 96;                // 32
    float* W2p = c1p + 32;                // 2048
    float* c2p = W2p + 2048;              // 64
    float* bcp = c2p + 64;                // 256
    float* Wcp = bcp + 256;               // 196608
    int*   idx = (int*)(Wcp + 196608);    // ROWS ints
    float* locl = (float*)(idx + ROWS);   // ROWS*3 floats
    float* X = locl + (size_t)ROWS * 3;   // BM*768 floats

    hipMemsetAsync(stats, 0, 512 * sizeof(float), stream);

    k_knn<<<dim3(MM / 256, BB), 256, 0, stream>>>(points, queries, idx, locl, stats);
    k_fold1<<<1, 32, 0, stream>>>(stats, g1, b1, W1, c1, W1p, c1p);
    k_bn2stats<<<256, 256, 0, stream>>>(locl, W1p, c1p, stats);
    k_fold2<<<1, 64, 0, stream>>>(stats, g2, b2, W2, c2, W2p, c2p);
    k_dense2pool<<<ROWS / 128, 256, 0, stream>>>(locl, W1p, c1p, W2p, c2p, X);
    k_prevpool<<<BM, 128, 0, stream>>>(prev, idx, X);
    k_bn3stats<<<64, 256, 0, stream>>>(X, stats);
    k_fold3<<<F2C, 256, 0, stream>>>(stats, g3, b3, Wc, bc, Wcp, bcp);
    k_gemm_out<<<BM / 16, 256, 0, stream>>>(X, Wcp, bcp, out);
}